// FreeYOLOv2_28424093565788
// MI455X (gfx1250) — compile-verified
//
#include <hip/hip_runtime.h>
#include <hip/hip_bf16.h>

typedef float v2f __attribute__((ext_vector_type(2)));
typedef float v8f __attribute__((ext_vector_type(8)));

#define KDIM 256
#define NUM_CLS 80
#define REG_CH 64
#define TOT_M 8400
#define OUT_CH 84

__device__ __forceinline__ v8f wmma_f32x4(v2f a, v2f b, v8f c) {
    // D(16x16,f32) = A(16x4,f32) * B(4x16,f32) + C
    return __builtin_amdgcn_wmma_f32_16x16x4_f32(
        /*neg_a=*/false, a, /*neg_b=*/false, b,
        /*c_mod=*/(short)0, c, /*reuse_a=*/false, /*reuse_b=*/false);
}

// ---------------------------------------------------------------------------
// Weight transpose + K-pair pack: in [O, 256] -> out [K/2][O][2]
// so a B tile (both K sub-rows for one lane) is ONE coalesced b64 load.
// out[((k/2)*O + o)*2 + (k&1)] = in[o*256 + k]
// ---------------------------------------------------------------------------
__global__ void yolo_wpack_kernel(const float* __restrict__ in,
                                  float* __restrict__ outP, int O) {
    int idx = blockIdx.x * blockDim.x + threadIdx.x;
    int total = O * KDIM;
    if (idx < total) {
        int j  = idx & 1;            // k & 1
        int oe = idx >> 1;           // (k/2)*O + o
        int p  = oe / O;             // k/2
        int o  = oe - p * O;
        outP[idx] = in[o * KDIM + 2 * p + j];
    }
}

// ---------------------------------------------------------------------------
// Fused per-level head: one wave = 16 positions x all 144 output channels.
//   cls GEMM (K=256, O=80, 5 N-tiles)  -> sigmoid
//   reg GEMM (K=256, O=64, 4 N-tiles)  -> DFL softmax + projection + box
// Level geometry is compile-time so all k-step feature offsets fold into
// the 24-bit instruction offsets (no per-step 64-bit address math).
// ---------------------------------------------------------------------------
template <int M, int WDIM, int LVL_OFF, int STRIDE>
__global__ __launch_bounds__(32)
void yolo_head_kernel(const float* __restrict__ cls_feat,  // [B,256,M]
                      const float* __restrict__ reg_feat,  // [B,256,M]
                      const float* __restrict__ clsWP,     // [128][80][2]
                      const float* __restrict__ regWP,     // [128][64][2]
                      const float* __restrict__ cls_b,     // [80]
                      const float* __restrict__ reg_b,     // [64]
                      float* __restrict__ out) {           // [B,8400,84]
    const int lane = threadIdx.x;        // 0..31
    const int half = lane >> 4;          // 0 or 1
    const int lr   = lane & 15;          // lane within 16-group
    const int mbase = blockIdx.x * 16;
    const int bidx  = blockIdx.y;
    const float stride_f = (float)STRIDE;

    const int m = mbase + lr;            // feature column this lane loads

    // Per-lane base pointers: the k index advances in compile-time constant
    // byte steps (k*M*4), which the compiler folds into immediate offsets.
    const float* __restrict__ cfp =
        cls_feat + (size_t)bidx * KDIM * M + (size_t)(2 * half) * M + m;
    const float* __restrict__ rfp =
        reg_feat + (size_t)bidx * KDIM * M + (size_t)(2 * half) * M + m;
    const v2f* __restrict__ wcP = (const v2f*)clsWP + (size_t)half * NUM_CLS + lr;
    const v2f* __restrict__ wrP = (const v2f*)regWP + (size_t)half * REG_CH + lr;

    v8f accC[5];
    v8f accR[4];
#pragma unroll
    for (int t = 0; t < 5; ++t)
#pragma unroll
        for (int v = 0; v < 8; ++v) accC[t][v] = 0.0f;
#pragma unroll
    for (int t = 0; t < 4; ++t)
#pragma unroll
        for (int v = 0; v < 8; ++v) accR[t][v] = 0.0f;

    // --- K loop: 64 steps of 4; 9 WMMAs, 4 NT feat loads, 9 b64 wgt loads ---
#pragma unroll 4
    for (int k0 = 0; k0 < KDIM; k0 += 4) {
        // A layout: lanes 0-15 hold K=k0,k0+1; lanes 16-31 hold K=k0+2,k0+3
        v2f aC, aR;
        aC.x = __builtin_nontemporal_load(&cfp[(size_t)k0 * M]);
        aC.y = __builtin_nontemporal_load(&cfp[(size_t)(k0 + 1) * M]);
        aR.x = __builtin_nontemporal_load(&rfp[(size_t)k0 * M]);
        aR.y = __builtin_nontemporal_load(&rfp[(size_t)(k0 + 1) * M]);

        // Weight K-pair row for this lane's half: index k0/2 (+half in base)
        const v2f* __restrict__ wc = wcP + (size_t)(k0 / 2) * NUM_CLS;
#pragma unroll
        for (int t = 0; t < 5; ++t) {
            v2f bb = wc[16 * t];                     // one b64 load
            accC[t] = wmma_f32x4(aC, bb, accC[t]);
        }
        const v2f* __restrict__ wr = wrP + (size_t)(k0 / 2) * REG_CH;
#pragma unroll
        for (int t = 0; t < 4; ++t) {
            v2f bb = wr[16 * t];                     // one b64 load
            accR[t] = wmma_f32x4(aR, bb, accR[t]);
        }
    }

    // --- Epilogue 1: class logits -> sigmoid, coalesced NT row stores ----
    const size_t rowbase = (size_t)bidx * TOT_M + LVL_OFF;
    float bias_c[5];
#pragma unroll
    for (int t = 0; t < 5; ++t) bias_c[t] = cls_b[16 * t + lr];

#pragma unroll
    for (int t = 0; t < 5; ++t) {
#pragma unroll
        for (int v = 0; v < 8; ++v) {
            const int mm = mbase + v + 8 * half;     // D layout row
            float x = accC[t][v] + bias_c[t];
            float s = 1.0f / (1.0f + __expf(-x));
            __builtin_nontemporal_store(
                s, &out[(rowbase + mm) * OUT_CH + 16 * t + lr]);
        }
    }

    // --- Epilogue 2: DFL softmax over 16 bins (16-lane butterfly) --------
    const float proj = (float)lr * (16.0f / 15.0f);  // linspace(0,16,16)[lr]
    float bias_r[4];
#pragma unroll
    for (int f = 0; f < 4; ++f) bias_r[f] = reg_b[16 * f + lr];

#pragma unroll
    for (int v = 0; v < 8; ++v) {
        float rv[4];
#pragma unroll
        for (int f = 0; f < 4; ++f) {
            float x = accR[f][v] + bias_r[f];
            float mx = x;
#pragma unroll
            for (int sh = 1; sh < 16; sh <<= 1)
                mx = fmaxf(mx, __shfl_xor(mx, sh));
            float e   = __expf(x - mx);
            float den = e;
            float num = e * proj;
#pragma unroll
            for (int sh = 1; sh < 16; sh <<= 1) {
                den += __shfl_xor(den, sh);
                num += __shfl_xor(num, sh);
            }
            rv[f] = num / den;           // expected bin value
        }
        const int mm = mbase + v + 8 * half;
        const float ax = ((float)(mm % WDIM) + 0.5f) * stride_f;
        const float ay = ((float)(mm / WDIM) + 0.5f) * stride_f;
        float bx;
        if      (lr == 0) bx = ax - rv[0] * stride_f;
        else if (lr == 1) bx = ay - rv[1] * stride_f;
        else if (lr == 2) bx = ax + rv[2] * stride_f;
        else              bx = ay + rv[3] * stride_f;
        if (lr < 4)
            __builtin_nontemporal_store(
                bx, &out[(rowbase + mm) * OUT_CH + NUM_CLS + lr]);
    }
}

// ---------------------------------------------------------------------------
extern "C" void kernel_launch(void* const* d_in, const int* in_sizes, int n_in,
                              void* d_out, int out_size, void* d_ws, size_t ws_size,
                              hipStream_t stream) {
    (void)in_sizes; (void)n_in; (void)out_size; (void)ws_size;
    // setup_inputs() dict order, per level i: cls_feat, reg_feat, cls_w, cls_b, reg_w, reg_b
    const float* cls_feat[3] = { (const float*)d_in[0], (const float*)d_in[6],  (const float*)d_in[12] };
    const float* reg_feat[3] = { (const float*)d_in[1], (const float*)d_in[7],  (const float*)d_in[13] };
    const float* cls_w[3]    = { (const float*)d_in[2], (const float*)d_in[8],  (const float*)d_in[14] };
    const float* cls_b[3]    = { (const float*)d_in[3], (const float*)d_in[9],  (const float*)d_in[15] };
    const float* reg_w[3]    = { (const float*)d_in[4], (const float*)d_in[10], (const float*)d_in[16] };
    const float* reg_b[3]    = { (const float*)d_in[5], (const float*)d_in[11], (const float*)d_in[17] };

    float* ws = (float*)d_ws;
    float* clsP[3]; float* regP[3];
    size_t off = 0;
    for (int l = 0; l < 3; ++l) { clsP[l] = ws + off; off += (size_t)KDIM * NUM_CLS; }
    for (int l = 0; l < 3; ++l) { regP[l] = ws + off; off += (size_t)KDIM * REG_CH; }

    // 1) transpose + K-pair pack weights into workspace (stream-ordered)
    for (int l = 0; l < 3; ++l) {
        yolo_wpack_kernel<<<(KDIM * NUM_CLS + 255) / 256, 256, 0, stream>>>(cls_w[l], clsP[l], NUM_CLS);
        yolo_wpack_kernel<<<(KDIM * REG_CH + 255) / 256, 256, 0, stream>>>(reg_w[l], regP[l], REG_CH);
    }

    // 2) fused per-level head kernels (compile-time geometry per level)
    const int B = 16;
    {   // level 0: 80x80, stride 8, M=6400, offset 0
        dim3 grid(6400 / 16, B);
        yolo_head_kernel<6400, 80, 0, 8><<<grid, 32, 0, stream>>>(
            cls_feat[0], reg_feat[0], clsP[0], regP[0], cls_b[0], reg_b[0], (float*)d_out);
    }
    {   // level 1: 40x40, stride 16, M=1600, offset 6400
        dim3 grid(1600 / 16, B);
        yolo_head_kernel<1600, 40, 6400, 16><<<grid, 32, 0, stream>>>(
            cls_feat[1], reg_feat[1], clsP[1], regP[1], cls_b[1], reg_b[1], (float*)d_out);
    }
    {   // level 2: 20x20, stride 32, M=400, offset 8000
        dim3 grid(400 / 16, B);
        yolo_head_kernel<400, 20, 8000, 32><<<grid, 32, 0, stream>>>(
            cls_feat[2], reg_feat[2], clsP[2], regP[2], cls_b[2], reg_b[2], (float*)d_out);
    }
}